// SparseLinear_51505247813854
// MI455X (gfx1250) — compile-verified
//
#include <hip/hip_runtime.h>

// fp32 WMMA operand/accumulator vector types (wave32)
typedef __attribute__((ext_vector_type(2))) float v2f;  // A/B operand: 2 VGPRs
typedef __attribute__((ext_vector_type(4))) float v4f;  // 16B load vector
typedef __attribute__((ext_vector_type(8))) float v8f;  // C/D: 16x16 f32 = 8 VGPRs

#define NB 256
#define NL 1000
#define ND 512
#define NTILES ((NL + 15) / 16)          // 63 L-tiles per batch
#define WAVES_PER_BLOCK 8                // 256 threads / wave32

__global__ __launch_bounds__(256) void sparse_linear_wmma_kernel(
    const float* __restrict__ embed,     // [B, D]
    const int*   __restrict__ shortlist, // [B, L]
    const float* __restrict__ weight,    // [V, D]
    const float* __restrict__ bias,      // [V, 1]
    float*       __restrict__ out)       // [B, L]
{
    const int lane = threadIdx.x & 31;
    const int wave = threadIdx.x >> 5;
    const int tile = blockIdx.x * WAVES_PER_BLOCK + wave;   // one wave per (b, l-tile)
    const int b  = tile / NTILES;
    const int lt = tile - b * NTILES;
    if (b >= NB) return;                                    // wave-uniform guard
    const int l0 = lt * 16;

    const int h = lane >> 4;       // half select: lanes 0-15 take k..k+3, lanes 16-31 take k+4..k+7
    const int m = lane & 15;       // matrix row within the tile

    int row = l0 + m;
    if (row >= NL) row = l0;       // clamp tail rows (loads stay in-bounds; stores masked later)

    const long long widx = (long long)shortlist[(long long)b * NL + row];
    const float* __restrict__ wrow = weight + widx * (long long)ND;   // gathered row (2 KB, cold HBM)
    const float* __restrict__ x    = embed + (long long)b * ND;       // hot in WGP$/L2

    v8f acc0 = {};
    v8f acc1 = {};

    // K loop: 512 / 8 = 64 iterations. One b128 weight load + one b128 embed load
    // per lane feeds TWO WMMAs: K-slot assignment inside a WMMA is arbitrary as
    // long as A and B pair the same physical k at the same slot, so lanes 0-15
    // contribute k..k+3 and lanes 16-31 contribute k+4..k+7 of each chunk.
    #pragma unroll 4
    for (int k = 0; k < ND; k += 8) {
        const int kb = k + 4 * h;
        v4f w4 = *(const v4f*)(wrow + kb);   // global_load_b128
        v4f x4 = *(const v4f*)(x + kb);      // global_load_b128 (broadcast per half)

        v2f wlo = __builtin_shufflevector(w4, w4, 0, 1);
        v2f xlo = __builtin_shufflevector(x4, x4, 0, 1);
        v2f whi = __builtin_shufflevector(w4, w4, 2, 3);
        v2f xhi = __builtin_shufflevector(x4, x4, 2, 3);

        acc0 = __builtin_amdgcn_wmma_f32_16x16x4_f32(
                   /*neg_a=*/false, wlo, /*neg_b=*/false, xlo,
                   /*c_mod=*/(short)0, acc0, /*reuse_a=*/false, /*reuse_b=*/false);
        acc1 = __builtin_amdgcn_wmma_f32_16x16x4_f32(
                   false, whi, false, xhi, (short)0, acc1, false, false);
    }
    v8f acc = acc0 + acc1;

    // D-matrix layout: VGPR j -> M=j (lanes 0-15), M=8+j (lanes 16-31); all N columns equal.
    // Lanes 0 and 16 each commit 8 rows (+ gathered bias), masking tail rows.
    if (m == 0) {
        const int baseRow = h * 8;
        #pragma unroll
        for (int j = 0; j < 8; ++j) {
            const int r = l0 + baseRow + j;
            if (r < NL) {
                const int vi = shortlist[(long long)b * NL + r];
                out[(long long)b * NL + r] = acc[j] + bias[vi];
            }
        }
    }
}

extern "C" void kernel_launch(void* const* d_in, const int* in_sizes, int n_in,
                              void* d_out, int out_size, void* d_ws, size_t ws_size,
                              hipStream_t stream) {
    const float* embed     = (const float*)d_in[0];   // [B, D]   fp32
    const int*   shortlist = (const int*)  d_in[1];   // [B, L]   int
    const float* weight    = (const float*)d_in[2];   // [V, D]   fp32
    const float* bias      = (const float*)d_in[3];   // [V, 1]   fp32
    float*       out       = (float*)d_out;           // [B, L]   fp32

    const int totalTiles = NB * NTILES;                       // 16128 waves
    const int blocks = (totalTiles + WAVES_PER_BLOCK - 1) / WAVES_PER_BLOCK;  // 2016
    sparse_linear_wmma_kernel<<<blocks, 256, 0, stream>>>(embed, shortlist, weight, bias, out);
}